// CapsuleLayer_52999896433077
// MI455X (gfx1250) — compile-verified
//
#include <hip/hip_runtime.h>
#include <math.h>

// Problem constants (from reference)
#define Bv 128    // batch
#define Pv 8      // primary capsule size (K of the GEMM)
#define Cv 1152   // input capsules
#define Uv 10     // output units
#define Sv 16     // unit size
#define Rv 160    // Uv*Sv  (M of the GEMM)

typedef __attribute__((ext_vector_type(2))) float v2f;
typedef __attribute__((ext_vector_type(8))) float v8f;

// ---------------------------------------------------------------------------
// Kernel 1: u_hat[c][r=u*16+s][b] = sum_p W[c,u,s,p] * X[b,p,c]
// Per c: [160 x 8] * [8 x 128] f32 GEMM via V_WMMA_F32_16X16X4_F32 (K=4, x2).
// One block of 128 threads (4 waves) per c; wave w handles N-tiles {2w,2w+1}.
// ---------------------------------------------------------------------------
__global__ void __launch_bounds__(128)
uhat_wmma_kernel(const float* __restrict__ X, const float* __restrict__ W,
                 float* __restrict__ uhat) {
    const int c    = blockIdx.x;
    const int wave = threadIdx.x >> 5;   // 0..3
    const int lane = threadIdx.x & 31;
    const int half = lane >> 4;          // 0 or 1
    const int l    = lane & 15;

    const float* Wc = W    + (size_t)c * (Rv * Pv);   // [160][8], contiguous in p
    float*       Uc = uhat + (size_t)c * (Rv * Bv);   // [160][128]

    for (int nt = wave * 2; nt < wave * 2 + 2; ++nt) {
        const int bcol = nt * 16 + l;                 // batch column of this lane
        // B-matrix (4x16, K x N): VGPR j holds K = k0 + 2*half + j, N = l
        const float* Xb = X + (size_t)bcol * (Pv * Cv) + c;
        v2f bf0, bf1;
        bf0.x = Xb[(size_t)(2 * half + 0) * Cv];
        bf0.y = Xb[(size_t)(2 * half + 1) * Cv];
        bf1.x = Xb[(size_t)(4 + 2 * half + 0) * Cv];
        bf1.y = Xb[(size_t)(4 + 2 * half + 1) * Cv];

        for (int mt = 0; mt < 10; ++mt) {
            // A-matrix (16x4, M x K): lane l holds M = mt*16+l, VGPR j -> K = 2*half + j
            const float* Wr = Wc + (size_t)(mt * 16 + l) * Pv;
            v2f af0, af1;
            af0.x = Wr[2 * half + 0];
            af0.y = Wr[2 * half + 1];
            af1.x = Wr[4 + 2 * half + 0];
            af1.y = Wr[4 + 2 * half + 1];

            v8f acc = {};
            acc = __builtin_amdgcn_wmma_f32_16x16x4_f32(
                false, af0, false, bf0, (short)0, acc, false, false);
            acc = __builtin_amdgcn_wmma_f32_16x16x4_f32(
                false, af1, false, bf1, (short)0, acc, false, false);

            // C/D layout: VGPR j -> M = mt*16 + j + 8*half, N = nt*16 + l
            float* Ucol = Uc + (size_t)(nt * 16 + l);
#pragma unroll
            for (int j = 0; j < 8; ++j) {
                Ucol[(size_t)(mt * 16 + j + 8 * half) * Bv] = acc[j];
            }
        }
    }
}

// ---------------------------------------------------------------------------
// Kernel 2: zero b_ij (workspace is poisoned by harness)
// ---------------------------------------------------------------------------
__global__ void zero_kernel(float* __restrict__ p, int n) {
    int i = blockIdx.x * blockDim.x + threadIdx.x;
    if (i < n) p[i] = 0.0f;
}

// ---------------------------------------------------------------------------
// Kernel 3: c_ij[:,u] = softmax over C of b_ij[:,u].  One block per u.
// ---------------------------------------------------------------------------
__global__ void __launch_bounds__(256)
softmax_kernel(const float* __restrict__ bij, float* __restrict__ cij) {
    const int u = blockIdx.x;
    __shared__ float red[256];
    const int tid = threadIdx.x;

    float m = -3.402823466e38f;
    for (int c = tid; c < Cv; c += 256) m = fmaxf(m, bij[c * Uv + u]);
    red[tid] = m; __syncthreads();
    for (int s = 128; s > 0; s >>= 1) {
        if (tid < s) red[tid] = fmaxf(red[tid], red[tid + s]);
        __syncthreads();
    }
    m = red[0]; __syncthreads();

    float sum = 0.0f;
    for (int c = tid; c < Cv; c += 256) sum += __expf(bij[c * Uv + u] - m);
    red[tid] = sum; __syncthreads();
    for (int s = 128; s > 0; s >>= 1) {
        if (tid < s) red[tid] += red[tid + s];
        __syncthreads();
    }
    const float inv = 1.0f / red[0];
    for (int c = tid; c < Cv; c += 256)
        cij[c * Uv + u] = __expf(bij[c * Uv + u] - m) * inv;
}

// ---------------------------------------------------------------------------
// Kernel 4: s_j[b,u,s] = sum_c c_ij[c,u] * uhat[c][u*16+s][b]
// One block per r = u*16+s (160 blocks), one thread per b (coalesced 512B/c).
// ---------------------------------------------------------------------------
__global__ void __launch_bounds__(128)
sj_kernel(const float* __restrict__ cij, const float* __restrict__ uhat,
          float* __restrict__ sj) {
    const int r = blockIdx.x;          // u*16 + s
    const int u = r >> 4;
    const int b = threadIdx.x;
    const float* up = uhat + (size_t)r * Bv + b;
    float acc = 0.0f;
#pragma unroll 4
    for (int c = 0; c < Cv; ++c) {
        acc += cij[c * Uv + u] * up[(size_t)c * (Rv * Bv)];
    }
    sj[b * Rv + r] = acc;
}

// ---------------------------------------------------------------------------
// Kernel 5: squash (note: reference squashes with norm over the U axis)
// v[b,u,s] = s_j[b,u,s] * sqrt(msq)/(1+msq), msq = sum_u s_j[b,u,s]^2
// ---------------------------------------------------------------------------
__global__ void squash_kernel(const float* __restrict__ sj, float* __restrict__ v) {
    const int t = blockIdx.x * blockDim.x + threadIdx.x;
    if (t >= Bv * Sv) return;
    const int b = t >> 4, s = t & 15;
    const float* sp = sj + b * Rv + s;
    float msq = 0.0f;
#pragma unroll
    for (int u = 0; u < Uv; ++u) { float x = sp[u * Sv]; msq += x * x; }
    const float scale = sqrtf(msq) / (1.0f + msq);
    float* vp = v + b * Rv + s;
#pragma unroll
    for (int u = 0; u < Uv; ++u) vp[u * Sv] = sp[u * Sv] * scale;
}

// ---------------------------------------------------------------------------
// Kernel 6: b_ij[c,u] += (1/B) * sum_{b,s} uhat[c][u*16+s][b] * v[b,u,s]
// One block per (c,u), 256-thread tree reduction over 16*128 elements.
// ---------------------------------------------------------------------------
__global__ void __launch_bounds__(256)
agree_kernel(const float* __restrict__ uhat, const float* __restrict__ v,
             float* __restrict__ bij) {
    const int cu = blockIdx.x;
    const int c = cu / Uv, u = cu % Uv;
    __shared__ float red[256];
    const int tid = threadIdx.x;
    const float* up = uhat + ((size_t)c * Rv + u * Sv) * Bv;  // [16][128]
    const float* vp = v + u * Sv;                             // v[b*160 + u*16 + s]
    float acc = 0.0f;
    for (int i = tid; i < Sv * Bv; i += 256) {
        const int s = i >> 7, b = i & 127;
        acc += up[s * Bv + b] * vp[b * Rv + s];
    }
    red[tid] = acc; __syncthreads();
    for (int st = 128; st > 0; st >>= 1) {
        if (tid < st) red[tid] += red[tid + st];
        __syncthreads();
    }
    if (tid == 0) bij[cu] += red[0] * (1.0f / Bv);
}

// ---------------------------------------------------------------------------
// Host launcher
// ---------------------------------------------------------------------------
extern "C" void kernel_launch(void* const* d_in, const int* in_sizes, int n_in,
                              void* d_out, int out_size, void* d_ws, size_t ws_size,
                              hipStream_t stream) {
    (void)in_sizes; (void)n_in; (void)out_size; (void)ws_size;
    const float* X = (const float*)d_in[0];   // [B,P,C]
    const float* W = (const float*)d_in[1];   // [C,U,S,P]
    float* out = (float*)d_out;               // [B,U,S,1]

    char* ws = (char*)d_ws;
    size_t off = 0;
    float* uhat = (float*)(ws + off); off += (size_t)Cv * Rv * Bv * sizeof(float); // 94.4 MB
    float* bij  = (float*)(ws + off); off += (size_t)Cv * Uv * sizeof(float);
    float* cij  = (float*)(ws + off); off += (size_t)Cv * Uv * sizeof(float);
    float* sj   = (float*)(ws + off); off += (size_t)Bv * Rv * sizeof(float);
    float* vbuf = (float*)(ws + off); off += (size_t)Bv * Rv * sizeof(float);

    // u_hat producer: 94 MB written once; stays resident in MI455X's 192MB L2
    // for all three routing passes below.
    uhat_wmma_kernel<<<Cv, 128, 0, stream>>>(X, W, uhat);
    zero_kernel<<<(Cv * Uv + 255) / 256, 256, 0, stream>>>(bij, Cv * Uv);

    for (int it = 0; it < 3; ++it) {
        softmax_kernel<<<Uv, 256, 0, stream>>>(bij, cij);
        sj_kernel<<<Rv, 128, 0, stream>>>(cij, uhat, sj);
        squash_kernel<<<(Bv * Sv + 255) / 256, 256, 0, stream>>>(
            sj, (it == 2) ? out : vbuf);
        if (it < 2) {  // final agree/b_ij update is dead in the reference
            agree_kernel<<<Cv * Uv, 256, 0, stream>>>(uhat, vbuf, bij);
        }
    }
}